// SPHERE_CUDA_41875931136706
// MI455X (gfx1250) — compile-verified
//
#include <hip/hip_runtime.h>
#include <hip/hip_bf16.h>

// Problem constants (from reference)
#define BB   4
#define CC   128
#define HH   128
#define WW   128
#define HWD  (HH * WW)         // 16384
#define SPH  32768
#define VV   32
#define PP   4096

typedef __attribute__((ext_vector_type(2))) float v2f;
typedef __attribute__((ext_vector_type(8))) float v8f;

// ---------------------------------------------------------------------------
// Kernel 1: transpose x [B, C, HW] -> xt [B, HW, C] so that one gathered vote
// (all 128 channels of one hw index) is a contiguous 512-byte row.
// ---------------------------------------------------------------------------
__global__ __launch_bounds__(256) void sphere_transpose(
    const float* __restrict__ x, float* __restrict__ xt)
{
    __shared__ float tile[32][33];   // +1 pad: no LDS bank conflicts
    const int b   = blockIdx.z;
    const int hw0 = blockIdx.x * 32;
    const int c0  = blockIdx.y * 32;
    const int tx  = threadIdx.x;     // 0..31
    const int ty  = threadIdx.y;     // 0..7

    const float* xb  = x  + (size_t)b * CC * HWD;
    float*       xtb = xt + (size_t)b * HWD * CC;

#pragma unroll
    for (int i = 0; i < 4; ++i)
        tile[ty + i * 8][tx] = xb[(size_t)(c0 + ty + i * 8) * HWD + hw0 + tx];

    __syncthreads();

#pragma unroll
    for (int i = 0; i < 4; ++i)
        xtb[(size_t)(hw0 + ty + i * 8) * CC + c0 + tx] = tile[tx][ty + i * 8];
}

// ---------------------------------------------------------------------------
// Kernel 2: weighted vote accumulation via V_WMMA_F32_16X16X4_F32.
//
// One workgroup = (batch b, tile of 16 points). 8 waves, wave w owns channel
// tile c in [16w, 16w+16).  For each point pn (0..15) and each chunk of 4
// votes: A[16c x 4v] = gathered x rows (exact f32), B[4 x 16] = zero except
// column pn, which carries that point's 4 vote weights.  Zero columns
// contribute exactly 0, so after 16*8 = 128 WMMAs the f32 accumulator
// D[16c x 16p] holds the exact fp32 result for the whole tile.
//
// Lane (hi*16+lh) only ever supplies a non-zero B value when pn == lh, so
// each lane preloads the 16 weights of point p = lh into registers once and
// builds B with a 0/1 mask multiply -> no per-chunk LDS reads, no exec-mask
// branching (pure v_cndmask + v_mul).
//
// SIDX/SC are compile-time strides so gather addressing is a single
// 32-bit lshl_add (per-batch x span is 8 MB -> 32-bit offsets are safe):
//   transposed xt[B,HW,C]: SIDX = C,  SC = 1   (fast path)
//   original    x[B,C,HW]: SIDX = 1,  SC = HW  (fallback)
// ---------------------------------------------------------------------------
template <int SIDX, int SC>
__global__ __launch_bounds__(256) void sphere_vote_wmma(
    const float* __restrict__ xsrc,
    const int* __restrict__ vote_index,
    const float* __restrict__ vote_weight,
    const long long* __restrict__ inds,
    float* __restrict__ out)
{
    const int wg = blockIdx.x;           // B * P/16 = 1024 workgroups
    const int b  = wg >> 8;              // 256 p-tiles per batch
    const int p0 = (wg & 255) << 4;

    __shared__ __align__(16) int   s_idx[16][VV];
    __shared__ __align__(16) float s_w[16][VV];

    const int t = threadIdx.x;
    for (int i = t; i < 16 * VV; i += 256) {
        const int p = i >> 5;
        const int v = i & 31;
        const long long srow = inds[(size_t)b * PP + p0 + p];
        s_idx[p][v] = vote_index[(size_t)srow * VV + v];
        s_w[p][v]   = vote_weight[(size_t)srow * VV + v];
    }
    __syncthreads();

    const int wave = t >> 5;             // 0..7 -> channel tile
    const int lane = t & 31;
    const int lh   = lane & 15;          // 0..15
    const int hi   = lane >> 4;          // 0 or 1
    const int c0   = wave << 4;

    // Preload this lane's own point's weights (only used when pn == lh).
    // Lane supplies B rows k = 2*hi and 2*hi+1 of each 4-vote chunk.
    v2f wown[8];
#pragma unroll
    for (int j = 0; j < 8; ++j) {
        const int kk = 4 * j + 2 * hi;
        wown[j].x = s_w[lh][kk];
        wown[j].y = s_w[lh][kk + 1];
    }

    const float* xb = xsrc + (size_t)b * CC * HWD;
    const int    cI = (c0 + lh) * SC;    // 32-bit channel offset

    v8f acc = {};                        // D accumulator, starts at 0

    for (int pn = 0; pn < 16; ++pn) {
        const float msk = (lh == pn) ? 1.0f : 0.0f;   // column select
#pragma unroll
        for (int j = 0; j < 8; ++j) {
            const int kk = 4 * j + 2 * hi;
            const int2 ii = *(const int2*)&s_idx[pn][kk];   // ds_load_b64

            // A-matrix 16x4 f32: lane (hi*16+lh) holds A[m=lh, k=2*hi] and
            // A[m=lh, k=2*hi+1]  (ISA 7.12.2 "32-bit A-Matrix 16x4")
            v2f a;
            a.x = xb[ii.x * SIDX + cI];
            a.y = xb[ii.y * SIDX + cI];

            // B-matrix 4x16 f32: VGPR0 row K=2*hi, VGPR1 row K=2*hi+1,
            // column n = lane%16.  Only column pn is non-zero.
            v2f bm;
            bm.x = wown[j].x * msk;
            bm.y = wown[j].y * msk;

            acc = __builtin_amdgcn_wmma_f32_16x16x4_f32(
                /*neg_a=*/false, a, /*neg_b=*/false, bm,
                /*c_mod=*/(short)0, acc, /*reuse_a=*/false, /*reuse_b=*/false);
        }
    }

    // D layout (ISA 7.12.2): VGPR r, lanes 0-15 -> M=r, lanes 16-31 -> M=r+8;
    // N = lane%16.  M = channel within tile, N = point within tile.
    float* ob = out + (size_t)b * CC * PP;
#pragma unroll
    for (int r = 0; r < 8; ++r) {
        const int c = c0 + r + 8 * hi;
        ob[(size_t)c * PP + p0 + lh] = acc[r];
    }
}

// ---------------------------------------------------------------------------
extern "C" void kernel_launch(void* const* d_in, const int* in_sizes, int n_in,
                              void* d_out, int out_size, void* d_ws, size_t ws_size,
                              hipStream_t stream)
{
    const float*     x           = (const float*)d_in[0];      // [B,C,H,W] f32
    const int*       vote_index  = (const int*)d_in[1];        // [S,V] i32
    const float*     vote_weight = (const float*)d_in[2];      // [S,V] f32
    const long long* inds        = (const long long*)d_in[3];  // [B,P] i64
    float*           out         = (float*)d_out;              // [B,C,P] f32

    const size_t xtBytes = (size_t)BB * HWD * CC * sizeof(float);  // 32 MB

    const dim3 voteGrid(BB * (PP / 16));    // 1024 WGs x 256 threads

    if (ws_size >= xtBytes) {
        // Fast path: transpose into scratch, then contiguous 512B gathers.
        float* xt = (float*)d_ws;
        dim3 tb(32, 8, 1);
        dim3 tg(HWD / 32, CC / 32, BB);
        sphere_transpose<<<tg, tb, 0, stream>>>(x, xt);
        sphere_vote_wmma<CC, 1><<<voteGrid, 256, 0, stream>>>(
            xt, vote_index, vote_weight, inds, out);
    } else {
        // Fallback: gather directly from [B,C,HW] (strided but correct).
        sphere_vote_wmma<1, HWD><<<voteGrid, 256, 0, stream>>>(
            x, vote_index, vote_weight, inds, out);
    }
}